// MMDLoss_1443109012023
// MI455X (gfx1250) — compile-verified
//
#include <hip/hip_runtime.h>
#include <hip/hip_bf16.h>

#define N_SRC 4096
#define DIM   256
#define NTOT  8192

typedef __attribute__((ext_vector_type(2))) float v2f;
typedef __attribute__((ext_vector_type(8))) float v8f;

// Sanity check strictly on the DEVICE pass: hipcc defines __AMDGCN__ even for
// the host pass, but __HIP_DEVICE_COMPILE__ is device-only.
#if defined(__HIP_DEVICE_COMPILE__) && !__has_builtin(__builtin_amdgcn_wmma_f32_16x16x4_f32)
#error "missing __builtin_amdgcn_wmma_f32_16x16x4_f32 on this toolchain (device pass)"
#endif

#define WMMA_F32X4(A, B, C) \
    __builtin_amdgcn_wmma_f32_16x16x4_f32(false, (A), false, (B), (short)0, (C), false, false)

// Scheduler pipeline hints: 0x020 = VMEM read group, 0x008 = MFMA/WMMA group.
#if defined(__HIP_DEVICE_COMPILE__) && __has_builtin(__builtin_amdgcn_sched_group_barrier)
#define SGB(mask, size, id) __builtin_amdgcn_sched_group_barrier((mask), (size), (id))
#else
#define SGB(mask, size, id)
#endif

// Row pointer into the virtual concat [source; target]
__device__ __forceinline__ const float* rowp(const float* __restrict__ s,
                                             const float* __restrict__ t, int i) {
    return (i < N_SRC) ? (s + (size_t)i * DIM) : (t + (size_t)(i - N_SRC) * DIM);
}

__device__ __forceinline__ float wave_sum(float v) {
    // wave32: full butterfly over 32 lanes
    for (int o = 16; o > 0; o >>= 1) v += __shfl_xor(v, o, 32);
    return v;
}

__device__ __forceinline__ void block_atomic_add(float v, float* dst) {
    __shared__ float red[8];
    const int lane = threadIdx.x & 31;
    const int w    = threadIdx.x >> 5;
    v = wave_sum(v);
    if (lane == 0) red[w] = v;
    __syncthreads();
    if (threadIdx.x == 0) {
        float s = 0.f;
        #pragma unroll
        for (int i = 0; i < 8; ++i) s += red[i];
        atomicAdd(dst, s);
    }
}

// One k-step worth of fragments: A(16x4) + 4x B(4x16)
struct Frag { v2f a, b0, b1, b2, b3; };

__device__ __forceinline__ Frag load_frag(const float* __restrict__ ra,
                                          const float* __restrict__ rb0,
                                          const float* __restrict__ rb1,
                                          const float* __restrict__ rb2,
                                          const float* __restrict__ rb3, int c) {
    Frag f;
    f.a  = *(const v2f*)(ra  + c);
    f.b0 = *(const v2f*)(rb0 + c);
    f.b1 = *(const v2f*)(rb1 + c);
    f.b2 = *(const v2f*)(rb2 + c);
    f.b3 = *(const v2f*)(rb3 + c);
    return f;
}

__device__ __forceinline__ void do_wmma(const Frag& f, v8f acc[4]) {
    acc[0] = WMMA_F32X4(f.a, f.b0, acc[0]);
    acc[1] = WMMA_F32X4(f.a, f.b1, acc[1]);
    acc[2] = WMMA_F32X4(f.a, f.b2, acc[2]);
    acc[3] = WMMA_F32X4(f.a, f.b3, acc[3]);
}

// One wave computes G strip: rows [i0,i0+16) x cols [j0,j0+64) of T*T^T,
// as 4 accumulators of v_wmma_f32_16x16x4_f32 over K=256.
// f32 16x16x4 layouts (wave32):
//   A (16x4, MxK): lanes 0-15 M=lane {K=0,1 in v0,v1}; lanes 16-31 M=lane-16 {K=2,3}
//   B (4x16, KxN): mirrored with N in place of M  -> identical gather pattern
//   C/D (16x16):   vgpr r, lane l -> (M = r + 8*(l>=16), N = l&15)
// Distance-2 software pipeline: iteration consumes k-steps {kk, kk+4} while
// prefetching {kk+8, kk+12} into two fresh fragment sets, so WMMAs only wait
// on loads issued two groups earlier. sched_group_barrier pins the schedule
// to [5xVMEM][4xWMMA][5xVMEM][4xWMMA] per iteration.
__device__ __forceinline__ void gemm_strip(const float* __restrict__ src,
                                           const float* __restrict__ tgt,
                                           int i0, int j0, int lane, v8f acc[4]) {
    const int m    = lane & 15;
    const int koff = (lane & 16) ? 2 : 0;
    const float* __restrict__ ra  = rowp(src, tgt, i0 + m) + koff;
    const float* __restrict__ rb0 = rowp(src, tgt, j0 +  0 + m) + koff;
    const float* __restrict__ rb1 = rowp(src, tgt, j0 + 16 + m) + koff;
    const float* __restrict__ rb2 = rowp(src, tgt, j0 + 32 + m) + koff;
    const float* __restrict__ rb3 = rowp(src, tgt, j0 + 48 + m) + koff;

    Frag s0 = load_frag(ra, rb0, rb1, rb2, rb3, 0);
    Frag s1 = load_frag(ra, rb0, rb1, rb2, rb3, 4);

    // kk = 0, 8, ..., 240 : 31 iterations, consuming k = 0..244, loading k = 8..252
    for (int kk = 0; kk + 12 <= DIM; kk += 8) {
        Frag s2 = load_frag(ra, rb0, rb1, rb2, rb3, kk + 8);
        do_wmma(s0, acc);
        Frag s3 = load_frag(ra, rb0, rb1, rb2, rb3, kk + 12);
        do_wmma(s1, acc);
        s0 = s2; s1 = s3;
        SGB(0x020, 5, 0);   // 5 VMEM reads  (prefetch set s2)
        SGB(0x008, 4, 0);   // 4 WMMA        (consume s0)
        SGB(0x020, 5, 0);   // 5 VMEM reads  (prefetch set s3)
        SGB(0x008, 4, 0);   // 4 WMMA        (consume s1)
    }
    // Epilogue: k = 248, 252
    do_wmma(s0, acc);
    do_wmma(s1, acc);
}

__global__ void k_init(float* __restrict__ w) {
    if (threadIdx.x < 2) w[threadIdx.x] = 0.f;
}

// sq[i] = ||T[i]||^2 ; one wave per row
__global__ __launch_bounds__(256) void k_rowsq(const float* __restrict__ src,
                                               const float* __restrict__ tgt,
                                               float* __restrict__ sq) {
    const int row  = blockIdx.x * 8 + (threadIdx.x >> 5);
    const int lane = threadIdx.x & 31;
    const float* __restrict__ r = rowp(src, tgt, row);
    float s = 0.f;
    #pragma unroll
    for (int c0 = 0; c0 < DIM; c0 += 32) { float x = r[c0 + lane]; s += x * x; }
    s = wave_sum(s);
    if (lane == 0) sq[row] = s;
}

// Pass 1: accumulate sum over all (i,j) of max(0, sq_i + sq_j - 2*G_ij)
__global__ __launch_bounds__(256) void k_sumd2(const float* __restrict__ src,
                                               const float* __restrict__ tgt,
                                               const float* __restrict__ sq,
                                               float* __restrict__ accum) {
    const int lane = threadIdx.x & 31;
    const int wv   = threadIdx.x >> 5;
    const int i0   = blockIdx.y * 128 + wv * 16;
    const int j0   = blockIdx.x * 64;
    v8f acc[4] = {};
    gemm_strip(src, tgt, i0, j0, lane, acc);

    const int hi8 = (lane & 16) ? 8 : 0;
    const int m   = lane & 15;
    float sqi[8];
    #pragma unroll
    for (int r = 0; r < 8; ++r) sqi[r] = sq[i0 + hi8 + r];

    float local = 0.f;
    #pragma unroll
    for (int jt = 0; jt < 4; ++jt) {
        const float sqj = sq[j0 + jt * 16 + m];
        #pragma unroll
        for (int r = 0; r < 8; ++r)
            local += fmaxf(sqi[r] + sqj - 2.0f * acc[jt][r], 0.0f);
    }
    block_atomic_add(local, accum);
}

// Pass 2: recompute G, evaluate 5-bandwidth Gaussian sum as e + e^2 + e^4 + e^8 + e^16
// with e = exp(-d2 / (16*bw)), signed-accumulate per region.
__global__ __launch_bounds__(256) void k_mmd(const float* __restrict__ src,
                                             const float* __restrict__ tgt,
                                             const float* __restrict__ sq,
                                             const float* __restrict__ sumd2,
                                             float* __restrict__ accum) {
    const int lane = threadIdx.x & 31;
    const int wv   = threadIdx.x >> 5;
    const int i0   = blockIdx.y * 128 + wv * 16;
    const int j0   = blockIdx.x * 64;
    v8f acc[4] = {};
    gemm_strip(src, tgt, i0, j0, lane, acc);

    // bandwidth = sum(d2)/(N^2-N) / 2^(5//2) ; largest bandwidth = 16*bw
    const float bw   = (*sumd2) * (1.0f / 67100672.0f) * 0.25f;
    const float ninv = -1.0f / (16.0f * bw);

    const int hi8 = (lane & 16) ? 8 : 0;
    const int m   = lane & 15;
    float sqi[8];
    #pragma unroll
    for (int r = 0; r < 8; ++r) sqi[r] = sq[i0 + hi8 + r];

    const bool iin = (i0 < N_SRC);       // 4096 is tile-aligned: uniform per tile
    float local = 0.f;
    #pragma unroll
    for (int jt = 0; jt < 4; ++jt) {
        const bool  jin = (j0 + jt * 16) < N_SRC;
        const float sgn = (iin == jin) ? 1.0f : -1.0f;
        const float sqj = sq[j0 + jt * 16 + m];
        float ts = 0.f;
        #pragma unroll
        for (int r = 0; r < 8; ++r) {
            const float d2  = fmaxf(sqi[r] + sqj - 2.0f * acc[jt][r], 0.0f);
            const float e   = __expf(d2 * ninv);     // v_exp_f32
            const float e2  = e  * e;
            const float e4  = e2 * e2;
            const float e8  = e4 * e4;
            const float e16 = e8 * e8;
            ts += e + e2 + e4 + e8 + e16;
        }
        local += sgn * ts;
    }
    block_atomic_add(local, accum);
}

__global__ void k_final(const float* __restrict__ accum, float* __restrict__ out) {
    if (threadIdx.x == 0 && blockIdx.x == 0)
        out[0] = accum[0] * (1.0f / ((float)N_SRC * (float)N_SRC));
}

extern "C" void kernel_launch(void* const* d_in, const int* in_sizes, int n_in,
                              void* d_out, int out_size, void* d_ws, size_t ws_size,
                              hipStream_t stream) {
    const float* src = (const float*)d_in[0];
    const float* tgt = (const float*)d_in[1];
    float* w       = (float*)d_ws;
    float* acc_d2  = w + 0;     // sum of d2
    float* acc_mmd = w + 1;     // signed kernel sum
    float* sq      = w + 8;     // 8192 row norms
    float* out     = (float*)d_out;

    k_init <<<1, 32, 0, stream>>>(w);
    k_rowsq<<<NTOT / 8, 256, 0, stream>>>(src, tgt, sq);
    dim3 grid(NTOT / 64, NTOT / 128);   // 128 j-tiles x 64 i-blocks
    k_sumd2<<<grid, 256, 0, stream>>>(src, tgt, sq, acc_d2);
    k_mmd  <<<grid, 256, 0, stream>>>(src, tgt, sq, acc_d2, acc_mmd);
    k_final<<<1, 1, 0, stream>>>(acc_mmd, out);
}